// TopKProtocol_62371515073182
// MI455X (gfx1250) — compile-verified
//
#include <hip/hip_runtime.h>
#include <stdint.h>

// Top-2-of-64 k-hot int64 mask, N = 1M rows.
// Streaming kernel: 256MB in + 512MB out @ 23.3 TB/s => ~33us floor.
// CDNA5 async global->LDS engine (ASYNCcnt), static double-buffered tiles,
// row-pair reduction (one row per 16-lane half-wave), scalar row-mask stores.

#define PATHS           64
#define ROWS_PER_TILE   8
#define TILE_FLOATS     (ROWS_PER_TILE * PATHS)     // 512 floats = 2048 B
#define TILE_BYTES      (TILE_FLOATS * 4)
#define WAVES_PER_BLOCK 8
#define BLOCK_THREADS   (WAVES_PER_BLOCK * 32)

struct __align__(16) ll2 { long long x, y; };

// One 8-row tile (2KB) = 4 x async b128 loads (32 lanes x 16B each).
// GV mode (64-bit vaddr, saddr=off), NT: single-pass stream.
__device__ __forceinline__ void issue_tile_async(const float* tile_lane_gptr,
                                                 unsigned lds_lane_addr) {
    uint64_t ga = (uint64_t)(uintptr_t)tile_lane_gptr;
    asm volatile(
        "global_load_async_to_lds_b128 %0, %1, off th:TH_LOAD_NT\n\t"
        "global_load_async_to_lds_b128 %0, %1, off offset:512 th:TH_LOAD_NT\n\t"
        "global_load_async_to_lds_b128 %0, %1, off offset:1024 th:TH_LOAD_NT\n\t"
        "global_load_async_to_lds_b128 %0, %1, off offset:1536 th:TH_LOAD_NT"
        :
        : "v"(lds_lane_addr), "v"(ga)
        : "memory");
}
__device__ __forceinline__ void wait_async_le4() {
    asm volatile("s_wait_asynccnt 0x4" ::: "memory");
}
__device__ __forceinline__ void wait_async_le0() {
    asm volatile("s_wait_asynccnt 0x0" ::: "memory");
}

// Monotonic sortable key: [63:32] = order-preserving f32 bits,
// [5:0] = 63 - col  (on value ties the SMALLER column wins the max,
// matching jax.lax.top_k). Excluded elements use key 0.
__device__ __forceinline__ uint64_t sort_key(float v, int col) {
    uint32_t u = __float_as_uint(v);
    uint32_t k = u ^ (uint32_t)(((int32_t)u >> 31) | 0x80000000);
    return ((uint64_t)k << 32) | (uint32_t)(PATHS - 1 - col);
}
__device__ __forceinline__ uint64_t umax64(uint64_t a, uint64_t b) {
    return (a > b) ? a : b;
}
// xor-reduce max within each 16-lane half (rows stay separated)
__device__ __forceinline__ uint64_t hmax16(uint64_t m) {
#pragma unroll
    for (int s = 8; s >= 1; s >>= 1)
        m = umax64(m, (uint64_t)__shfl_xor((unsigned long long)m, s, 32));
    return m;
}
// full-wave reduce (tail path only)
__device__ __forceinline__ uint64_t hmax32(uint64_t m) {
#pragma unroll
    for (int s = 16; s >= 1; s >>= 1)
        m = umax64(m, (uint64_t)__shfl_xor((unsigned long long)m, s, 32));
    return m;
}

// Emit one row's 64 int64 mask values: 2 bits per lane from a scalar mask.
__device__ __forceinline__ void store_row_mask(uint64_t smask, int lane,
                                               long long* __restrict__ orow) {
    const unsigned b = (unsigned)(smask >> (unsigned)(lane * 2)); // v_lshrrev_b64
    ll2 o;
    o.x = (long long)(b & 1u);
    o.y = (long long)((b >> 1) & 1u);
    *(ll2*)(orow + lane * 2) = o;                  // global_store_b128, 512B/row
}

// Two rows at once: lanes 0-15 hold row0 (4 cols/lane), lanes 16-31 row1.
__device__ __forceinline__ void process_pair(const float* __restrict__ pair_lds,
                                             int lane,
                                             long long* __restrict__ orow0) {
    const int h = lane >> 4;          // which row of the pair
    const int q = lane & 15;          // 16B chunk within the row
    float4 v = *(const float4*)(pair_lds + h * PATHS + q * 4); // ds_load_b128
    const int c0 = q * 4;

    const uint64_t k0 = sort_key(v.x, c0);
    const uint64_t k1 = sort_key(v.y, c0 + 1);
    const uint64_t k2 = sort_key(v.z, c0 + 2);
    const uint64_t k3 = sort_key(v.w, c0 + 3);

    uint64_t m = umax64(umax64(k0, k1), umax64(k2, k3));
    m = hmax16(m);
    const int top0 = (PATHS - 1) - (int)((uint32_t)m & 63u);

    const uint64_t e0 = (c0     == top0) ? 0ull : k0;
    const uint64_t e1 = (c0 + 1 == top0) ? 0ull : k1;
    const uint64_t e2 = (c0 + 2 == top0) ? 0ull : k2;
    const uint64_t e3 = (c0 + 3 == top0) ? 0ull : k3;
    uint64_t m2 = umax64(umax64(e0, e1), umax64(e2, e3));
    m2 = hmax16(m2);
    const int top1 = (PATHS - 1) - (int)((uint32_t)m2 & 63u);

    // winners -> SGPRs; scalar 64-bit row masks (s_lshl_b64/s_or_b64)
    const int t0r0 = __builtin_amdgcn_readlane(top0, 0);
    const int t1r0 = __builtin_amdgcn_readlane(top1, 0);
    const int t0r1 = __builtin_amdgcn_readlane(top0, 16);
    const int t1r1 = __builtin_amdgcn_readlane(top1, 16);
    const uint64_t mask0 = (1ull << t0r0) | (1ull << t1r0);
    const uint64_t mask1 = (1ull << t0r1) | (1ull << t1r1);

    store_row_mask(mask0, lane, orow0);
    store_row_mask(mask1, lane, orow0 + PATHS);
}

__device__ __forceinline__ void process_tile(const float* __restrict__ buf,
                                             int row_base, int lane,
                                             long long* __restrict__ out) {
#pragma unroll
    for (int p = 0; p < ROWS_PER_TILE / 2; ++p)
        process_pair(buf + p * 2 * PATHS, lane,
                     out + (size_t)(row_base + p * 2) * PATHS);
}

// tail (<8 rows): one row per whole wave, float2/lane
__device__ __forceinline__ void process_row_tail(float va, float vb, int lane,
                                                 long long* __restrict__ orow) {
    const int i0 = lane * 2, i1 = i0 + 1;
    const uint64_t k0 = sort_key(va, i0);
    const uint64_t k1 = sort_key(vb, i1);
    uint64_t m = hmax32(umax64(k0, k1));
    const int top0 = (PATHS - 1) - (int)((uint32_t)m & 63u);
    const uint64_t e0 = (i0 == top0) ? 0ull : k0;
    const uint64_t e1 = (i1 == top0) ? 0ull : k1;
    uint64_t m2 = hmax32(umax64(e0, e1));
    const int top1 = (PATHS - 1) - (int)((uint32_t)m2 & 63u);
    const int t0 = __builtin_amdgcn_readlane(top0, 0);
    const int t1 = __builtin_amdgcn_readlane(top1, 0);
    store_row_mask((1ull << t0) | (1ull << t1), lane, orow);
}

__global__ void __launch_bounds__(BLOCK_THREADS, 2)
topk2_khot_kernel(const float* __restrict__ score,
                  long long* __restrict__ out, int n, int tpw) {
    // per-wave static double buffer: 2 x 2KB; 8 waves -> 32KB of 320KB/WGP
    __shared__ float lds[WAVES_PER_BLOCK * 2 * TILE_FLOATS];

    const int lane  = threadIdx.x & 31;
    const int wave  = threadIdx.x >> 5;
    const int gwave = blockIdx.x * WAVES_PER_BLOCK + wave;

    const int full = n >> 3;          // full 8-row tiles
    const int tail = n & 7;

    // contiguous tile range per wave (tpw precomputed on host)
    const int begin = gwave * tpw;
    const int end   = (begin + tpw < full) ? (begin + tpw) : full;

    float* A = lds + wave * (2 * TILE_FLOATS);
    float* B = A + TILE_FLOATS;
    const unsigned a_addr = (unsigned)(uintptr_t)A + (unsigned)(lane * 16);
    const unsigned b_addr = a_addr + TILE_BYTES;

    int t = begin;
    if (t < end) {
        issue_tile_async(score + (size_t)t * TILE_FLOATS + lane * 4, a_addr);
        if (t + 1 < end)
            issue_tile_async(score + (size_t)(t + 1) * TILE_FLOATS + lane * 4, b_addr);

        while (t < end) {
            const bool hasB = (t + 1) < end;
            // ---- A: tile t ----  (in flight, oldest first: A(t)[, B(t+1)])
            if (hasB) wait_async_le4(); else wait_async_le0();
            process_tile(A, t * ROWS_PER_TILE, lane, out);
            if (t + 2 < end)   // A fully consumed; keep pipeline 2-deep
                issue_tile_async(score + (size_t)(t + 2) * TILE_FLOATS + lane * 4,
                                 a_addr);
            if (!hasB) break;
            // ---- B: tile t+1 ---- (in flight: B(t+1)[, A(t+2)])
            if (t + 2 < end) wait_async_le4(); else wait_async_le0();
            process_tile(B, (t + 1) * ROWS_PER_TILE, lane, out);
            if (t + 3 < end)
                issue_tile_async(score + (size_t)(t + 3) * TILE_FLOATS + lane * 4,
                                 b_addr);
            t += 2;
        }
    }

    // tail rows (<8): wave 0 only, direct global reads
    if (tail && gwave == 0) {
        for (int r = 0; r < tail; ++r) {
            const int row = full * ROWS_PER_TILE + r;
            float2 v = *(const float2*)(score + (size_t)row * PATHS + lane * 2);
            process_row_tail(v.x, v.y, lane, out + (size_t)row * PATHS);
        }
    }
    // S_ENDPGM implicitly waits idle for any still-outstanding async loads.
}

extern "C" void kernel_launch(void* const* d_in, const int* in_sizes, int n_in,
                              void* d_out, int out_size, void* d_ws, size_t ws_size,
                              hipStream_t stream) {
    (void)n_in; (void)d_ws; (void)ws_size; (void)out_size;
    const float* score = (const float*)d_in[0];
    long long*   out   = (long long*)d_out;     // reference output dtype: int64
    const int n = in_sizes[0] / PATHS;          // 1,048,576 rows

    const int full = n >> 3;
    int blocks = 2048;                          // 16K waves, 8 tiles each (N=1M)
    const int maxb = (full + WAVES_PER_BLOCK - 1) / WAVES_PER_BLOCK;
    if (blocks > maxb) blocks = maxb;
    if (blocks < 1) blocks = 1;

    const int nwave = blocks * WAVES_PER_BLOCK;
    const int tpw   = (full + nwave - 1) / nwave;   // host-side: no device idiv

    topk2_khot_kernel<<<blocks, BLOCK_THREADS, 0, stream>>>(score, out, n, tpw);
}